// LocallyConnected2D_40183714021732
// MI455X (gfx1250) — compile-verified
//
#include <hip/hip_runtime.h>

typedef __attribute__((ext_vector_type(2))) float v2f;
typedef __attribute__((ext_vector_type(8))) float v8f;

#define HH       64
#define WW       64
#define CIN      32
#define COUT     64
#define DEPTH    288          // CIN*3*3
#define S_TOT    4096         // HO*WO
#define BATCH    16
#define LDS_STR  292          // padded row stride (floats): b*292*? -> conflict-free b64 reads
#define THREADS  128          // 4 waves (wave32)

// One workgroup per spatial site s. Computes out[0:16, wave*16:+16, s] via
// D(16x16) += A(16x4) x B(4x16) WMMA chain over K=288.
__global__ __launch_bounds__(THREADS)
void lc2d_wmma_f32_kernel(const float* __restrict__ x,     // (16, 32, 64, 64)
                          const float* __restrict__ w,     // (288, 4096, 64)
                          const float* __restrict__ bias,  // (64, 64, 64) = (o, HO, WO)
                          float* __restrict__ out)         // (16, 64, 64, 64)
{
    __shared__ float patch[BATCH * LDS_STR];   // 18688 B: unfolded A, layout [b][d]

    const int s   = blockIdx.x;        // spatial site 0..4095
    const int oh  = s >> 6;
    const int ow  = s & 63;
    const int tid = threadIdx.x;

    // ---- Phase 1: stage 16x288 unfolded patch into LDS (shared A for all 4 waves)
    #pragma unroll
    for (int i = 0; i < (BATCH * DEPTH) / THREADS; ++i) {
        const int flat = i * THREADS + tid;       // = b*288 + d
        const int b  = flat / DEPTH;
        const int d  = flat - b * DEPTH;
        const int c  = d / 9;
        const int rr = d - c * 9;
        const int kh = rr / 3;
        const int kw = rr - kh * 3;
        const int ih = oh + kh - 1;               // pad=1, stride=1, dil=1
        const int iw = ow + kw - 1;
        float v = 0.0f;
        if ((unsigned)ih < (unsigned)HH && (unsigned)iw < (unsigned)WW)
            v = x[((b * CIN + c) * HH + ih) * WW + iw];
        patch[b * LDS_STR + d] = v;
    }
    __syncthreads();

    // ---- Phase 2: per-wave 16x16 tile of D over K=288 in steps of 4
    const int lane  = tid & 31;
    const int wave  = tid >> 5;
    const int l15   = lane & 15;       // A: row M; B: col N
    const int khalf = (lane >> 4) * 2; // 0 (lanes 0-15) or 2 (lanes 16-31)
    const int nbase = wave * 16;

    // A fragment source: patch[M][k0 + khalf + {0,1}], 8B-aligned -> ds_load_b64
    const float* ap = patch + l15 * LDS_STR + khalf;
    // B fragment source: W[k][s][n], n = nbase+l15 fixed per lane, k strided by S*COUT
    const float* wp = w + (unsigned)s * COUT + (unsigned)(nbase + l15);

    v8f acc = {};
    #pragma unroll 4
    for (int k0 = 0; k0 < DEPTH; k0 += 4) {
        const v2f a = *(const v2f*)(ap + k0);
        const unsigned kr = (unsigned)(k0 + khalf);
        v2f bfrag;
        // weights are a one-shot 302MB stream: keep them non-temporal (TH=NT)
        bfrag.x = __builtin_nontemporal_load(wp + (size_t)kr       * (S_TOT * COUT));
        bfrag.y = __builtin_nontemporal_load(wp + (size_t)(kr + 1) * (S_TOT * COUT));
        // D = A(16x4) * B(4x16) + C  -> v_wmma_f32_16x16x4_f32
        acc = __builtin_amdgcn_wmma_f32_16x16x4_f32(
            /*neg_a=*/false, a, /*neg_b=*/false, bfrag,
            /*c_mod=*/(short)0, acc, /*reuse_a=*/false, /*reuse_b=*/false);
    }

    // ---- Phase 3: bias + writeback.  D layout: VGPR r -> M = r + 8*(lane>=16), N = l15.
    const int o   = nbase + l15;                 // output channel (fixed per lane)
    const float bv = bias[(unsigned)o * S_TOT + (unsigned)s];
    const int mb  = (lane >> 4) * 8;
    float* op = out + ((unsigned)o * S_TOT + (unsigned)s);
    #pragma unroll
    for (int r = 0; r < 8; ++r) {
        const unsigned b = (unsigned)(mb + r);   // batch index = M
        __builtin_nontemporal_store(acc[r] + bv, op + (size_t)b * (COUT * S_TOT));
    }
}

extern "C" void kernel_launch(void* const* d_in, const int* in_sizes, int n_in,
                              void* d_out, int out_size, void* d_ws, size_t ws_size,
                              hipStream_t stream) {
    const float* x    = (const float*)d_in[0];   // (16,32,64,64) fp32
    const float* wgt  = (const float*)d_in[1];   // (1,288,4096,64) fp32
    const float* bias = (const float*)d_in[2];   // (1,64,64,64) fp32
    float* out        = (float*)d_out;           // (16,64,64,64) fp32

    lc2d_wmma_f32_kernel<<<S_TOT, THREADS, 0, stream>>>(x, wgt, bias, out);
}